// nonlocal_attention_798863917788
// MI455X (gfx1250) — compile-verified
//
#include <hip/hip_runtime.h>
#include <hip/hip_bf16.h>

typedef __attribute__((ext_vector_type(2))) float v2f;
typedef __attribute__((ext_vector_type(8))) float v8f;

#define HW 9216      // 96*96
#define CH 128
#define IC 64
#define NT 7
#define NB 2
#define NIMG 14      // NB*NT

// ---------------------------------------------------------------------------
// GroupNorm stats: one block per (image, group). Group = 4 contiguous channels
// of 9216 floats each -> fully coalesced 144KB reduction per block.
// ---------------------------------------------------------------------------
__global__ __launch_bounds__(256) void k_gn_stats(const float* __restrict__ x,
                                                  float* __restrict__ stats) {
  const int n = blockIdx.x >> 5;   // image 0..13
  const int g = blockIdx.x & 31;   // group 0..31
  const float* p = x + ((long)n * CH + g * 4) * HW;
  float s = 0.f, ss = 0.f;
  for (int i = threadIdx.x; i < 4 * HW; i += 256) {
    float v = p[i]; s += v; ss += v * v;
  }
  __shared__ float b1[256], b2[256];
  b1[threadIdx.x] = s; b2[threadIdx.x] = ss; __syncthreads();
  for (int st = 128; st > 0; st >>= 1) {
    if (threadIdx.x < st) { b1[threadIdx.x] += b1[threadIdx.x + st];
                            b2[threadIdx.x] += b2[threadIdx.x + st]; }
    __syncthreads();
  }
  if (threadIdx.x == 0) {
    const float invn = 1.f / 36864.f;
    float m = b1[0] * invn;
    float var = b2[0] * invn - m * m;
    stats[blockIdx.x * 2]     = m;
    stats[blockIdx.x * 2 + 1] = rsqrtf(var + 1e-6f);
  }
}

// ---------------------------------------------------------------------------
// Apply GroupNorm (channel-first, same layout as x). q_ == ht at t=3 slices.
// ---------------------------------------------------------------------------
__global__ __launch_bounds__(256) void k_gn_apply(const float* __restrict__ x,
    const float* __restrict__ stats, const float* __restrict__ gamma,
    const float* __restrict__ beta, float* __restrict__ ht, long total) {
  long i = (long)blockIdx.x * blockDim.x + threadIdx.x;
  const long stride = (long)gridDim.x * blockDim.x;
  for (; i < total; i += stride) {
    long cimg = i / HW;
    int c = (int)(cimg % CH);
    int n = (int)(cimg / CH);
    int sg = n * 32 + (c >> 2);
    float m = stats[sg * 2], r = stats[sg * 2 + 1];
    ht[i] = (x[i] - m) * r * gamma[c] + beta[c];
  }
}

// ---------------------------------------------------------------------------
// mbT[m,c] = mb[c,m]   (64x256 -> 256x64), tiny
// ---------------------------------------------------------------------------
__global__ __launch_bounds__(256) void k_mb_transpose(const float* __restrict__ mb,
                                                      float* __restrict__ mbT) {
  int i = blockIdx.x * 256 + threadIdx.x;   // 0..16383
  int c = i >> 8, m = i & 255;
  mbT[m * IC + c] = mb[i];
}

// ---------------------------------------------------------------------------
// WMMA fp32 GEMM: C[M x 9216] = alpha * W[M x K] @ B[K x 9216] + bias[M]
// One wave computes a 16x16 D tile via V_WMMA_F32_16X16X4_F32; 8 waves/block
// cover 128 columns. blockIdx.z selects the image (strideB / strideC).
// A lane: W[mrow0+l16, 2*lh + {0,1}]      (8B contiguous per lane)
// B lane: B[2*lh + {0,1}, ncol]           (64B contiguous per half-wave row)
// D VGPR r -> row mrow0 + r + 8*lh, col ncol (coalesced channel-first store)
// ---------------------------------------------------------------------------
__global__ __launch_bounds__(256) void k_gemm(const float* __restrict__ W,
    const float* __restrict__ Bbase, float* __restrict__ Cbase,
    const float* __restrict__ bias, int M, int K,
    long strideB, long strideC, float alpha) {
  const int N = HW;
  const float* B = Bbase + (long)blockIdx.z * strideB;
  float* C = Cbase + (long)blockIdx.z * strideC;
  const int lane = threadIdx.x & 31;
  const int wave = threadIdx.x >> 5;
  const int l16 = lane & 15;
  const int lh  = lane >> 4;
  const int ncol  = blockIdx.x * 128 + wave * 16 + l16;
  const int mrow0 = blockIdx.y * 16;
  const int arow  = mrow0 + l16;

  v8f acc = {0.f, 0.f, 0.f, 0.f, 0.f, 0.f, 0.f, 0.f};
  const float* wp = W + (long)arow * K + 2 * lh;
  const float* bp = B + (long)(2 * lh) * N + ncol;
  for (int k0 = 0; k0 < K; k0 += 4) {
    v2f a, b;
    a.x = wp[0]; a.y = wp[1];
    b.x = bp[0]; b.y = bp[N];
    acc = __builtin_amdgcn_wmma_f32_16x16x4_f32(false, a, false, b,
                                                (short)0, acc, false, false);
    wp += 4;
    bp += 4L * N;
  }
#pragma unroll
  for (int r = 0; r < 8; ++r) {
    int row = mrow0 + 8 * lh + r;
    float v = acc[r] * alpha;
    if (bias) v += bias[row];
    C[(long)row * N + ncol] = v;
  }
}

// ---------------------------------------------------------------------------
// 3x3 neighborhood attention. Thread = one spatial position. q/k/v are
// channel-first (coalesced loads along p). Output f written channel-LAST in
// (t, p, c) order via LDS staging (so the W_y GEMM can reinterpret flat f as
// a row-major 448 x 9216 B matrix, exactly matching the reference reshape).
// ---------------------------------------------------------------------------
__global__ __launch_bounds__(256) void k_attn(const float* __restrict__ phiT,
    const float* __restrict__ thetaT, const float* __restrict__ gT,
    float* __restrict__ f) {
  const int b = blockIdx.z, t = blockIdx.y;
  const int p0 = blockIdx.x * 256;
  const int p  = p0 + threadIdx.x;
  const int row = p / 96, col = p % 96;
  const int r0 = min(max(row - 1, 0), 93);
  const int c0 = min(max(col - 1, 0), 93);

  const float* q  = phiT + (long)b * IC * HW;
  const long img  = (long)(b * NT + t) * IC * HW;
  const float* kk = thetaT + img;
  const float* vv = gT + img;

  int np[9];
#pragma unroll
  for (int i = 0; i < 3; ++i)
#pragma unroll
    for (int j = 0; j < 3; ++j) np[i * 3 + j] = (r0 + i) * 96 + (c0 + j);

  float lg[9] = {0.f,0.f,0.f,0.f,0.f,0.f,0.f,0.f,0.f};
  for (int c = 0; c < IC; ++c) {
    float qc = q[(long)c * HW + p] * 0.125f;   // ic^-0.5
#pragma unroll
    for (int j = 0; j < 9; ++j) lg[j] += qc * kk[(long)c * HW + np[j]];
  }
  float mx = lg[0];
#pragma unroll
  for (int j = 1; j < 9; ++j) mx = fmaxf(mx, lg[j]);
  float s = 0.f;
#pragma unroll
  for (int j = 0; j < 9; ++j) { lg[j] = __expf(lg[j] - mx); s += lg[j]; }
  float inv = 1.f / s;
#pragma unroll
  for (int j = 0; j < 9; ++j) lg[j] *= inv;

  __shared__ float sf[256 * 17];   // 16-ch chunk, pad 17 -> bank-conflict free
  float* fb = f + (long)(b * NT + t) * (long)HW * IC;
  for (int cb = 0; cb < IC; cb += 16) {
#pragma unroll
    for (int cc = 0; cc < 16; ++cc) {
      int c = cb + cc;
      float o = 0.f;
#pragma unroll
      for (int j = 0; j < 9; ++j) o += lg[j] * vv[(long)c * HW + np[j]];
      sf[threadIdx.x * 17 + cc] = o;
    }
    __syncthreads();
#pragma unroll
    for (int qd = 0; qd < 4; ++qd) {
      int idx  = threadIdx.x + qd * 256;    // 0..1023
      int pos  = idx >> 2;
      int part = idx & 3;
      const float* src = &sf[pos * 17 + part * 4];
      float4 val = make_float4(src[0], src[1], src[2], src[3]);
      *(float4*)(fb + ((long)(p0 + pos)) * IC + cb + part * 4) = val;
    }
    __syncthreads();
  }
}

// ---------------------------------------------------------------------------
// Softmax over 256 memory entries; logitsT layout (b, 256, 9216), in place.
// Thread = one position p; the m-loop strides 9216 => coalesced across lanes.
// ---------------------------------------------------------------------------
__global__ __launch_bounds__(256) void k_softmax256(float* __restrict__ L0) {
  const int b = blockIdx.y;
  const int p = blockIdx.x * 256 + threadIdx.x;
  float* L = L0 + (long)b * 256 * HW;
  float mx = -3.4e38f;
  for (int m = 0; m < 256; ++m) mx = fmaxf(mx, L[(long)m * HW + p]);
  float s = 0.f;
  for (int m = 0; m < 256; ++m) s += __expf(L[(long)m * HW + p] - mx);
  float inv = 1.f / s;
  for (int m = 0; m < 256; ++m) {
    long i = (long)m * HW + p;
    L[i] = __expf(L[i] - mx) * inv;
  }
}

// ---------------------------------------------------------------------------
// qloss = mean |phi - y1| : deterministic two-stage reduction (no FP atomics)
// ---------------------------------------------------------------------------
__global__ __launch_bounds__(256) void k_l1_partial(const float* __restrict__ a,
    const float* __restrict__ b, float* __restrict__ part, long n) {
  float s = 0.f;
  for (long i = (long)blockIdx.x * 256 + threadIdx.x; i < n;
       i += 256L * gridDim.x)
    s += fabsf(a[i] - b[i]);
  __shared__ float sb[256];
  sb[threadIdx.x] = s; __syncthreads();
  for (int st = 128; st > 0; st >>= 1) {
    if (threadIdx.x < st) sb[threadIdx.x] += sb[threadIdx.x + st];
    __syncthreads();
  }
  if (threadIdx.x == 0) part[blockIdx.x] = sb[0];
}

__global__ __launch_bounds__(256) void k_l1_final(const float* __restrict__ part,
    int nb, float* __restrict__ out, float scale) {
  float s = 0.f;
  for (int i = threadIdx.x; i < nb; i += 256) s += part[i];
  __shared__ float sb[256];
  sb[threadIdx.x] = s; __syncthreads();
  for (int st = 128; st > 0; st >>= 1) {
    if (threadIdx.x < st) sb[threadIdx.x] += sb[threadIdx.x + st];
    __syncthreads();
  }
  if (threadIdx.x == 0) out[0] = sb[0] * scale;
}

// ---------------------------------------------------------------------------
// z = W_y + x[:,3] + W_y1   (channel-first, straight into d_out)
// ---------------------------------------------------------------------------
__global__ __launch_bounds__(256) void k_final_add(const float* __restrict__ Wy,
    const float* __restrict__ Wy1, const float* __restrict__ x,
    float* __restrict__ z, long total) {
  long i = (long)blockIdx.x * blockDim.x + threadIdx.x;
  const long stride = (long)gridDim.x * blockDim.x;
  const long per_b = (long)CH * HW;
  for (; i < total; i += stride) {
    long b = i / per_b, rem = i % per_b;
    float qv = x[(long)(b * NT + 3) * per_b + rem];
    z[i] = Wy[i] + Wy1[i] + qv;
  }
}

// ---------------------------------------------------------------------------
extern "C" void kernel_launch(void* const* d_in, const int* in_sizes, int n_in,
                              void* d_out, int out_size, void* d_ws, size_t ws_size,
                              hipStream_t stream) {
  (void)in_sizes; (void)n_in; (void)out_size; (void)ws_size;
  const float* x       = (const float*)d_in[0];
  const float* g_w     = (const float*)d_in[1];
  const float* g_b     = (const float*)d_in[2];
  const float* theta_w = (const float*)d_in[3];
  const float* theta_b = (const float*)d_in[4];
  const float* phi_w   = (const float*)d_in[5];
  const float* phi_b   = (const float*)d_in[6];
  const float* wz_w    = (const float*)d_in[7];
  const float* wz_b    = (const float*)d_in[8];
  const float* wz1_w   = (const float*)d_in[9];
  const float* wz1_b   = (const float*)d_in[10];
  const float* gamma   = (const float*)d_in[11];
  const float* beta    = (const float*)d_in[12];
  const float* mb      = (const float*)d_in[13];
  float* out = (float*)d_out;
  float* ws  = (float*)d_ws;

  // workspace layout (floats)
  float* stats  = ws;                      // 1024  (mean,rstd per image*group)
  float* mbT    = ws + 1024;               // 16384
  float* part   = ws + 1024 + 16384;       // 1024
  float* ht     = ws + 18432;              // 14*128*9216 = 16,515,072
  float* thetaT = ht + 16515072L;          // 14*64*9216  =  8,257,536
  float* gT     = thetaT + 8257536L;       //               8,257,536
  float* phiT   = gT + 8257536L;           // 2*64*9216   =  1,179,648
  float* f      = phiT + 1179648L;         // 2*7*9216*64 =  8,257,536
  float* Wy     = f + 8257536L;            // 2*128*9216  =  2,359,296
  float* lgts   = Wy + 2359296L;           // 2*256*9216  =  4,718,592
  float* y1T    = lgts + 4718592L;         // 2*64*9216   =  1,179,648
  float* Wy1    = y1T + 1179648L;          // 2*128*9216  =  2,359,296

  const long hw = HW;

  // 1) GroupNorm stats + apply (q_ == ht slices at t=3)
  k_gn_stats<<<NIMG * 32, 256, 0, stream>>>(x, stats);
  k_gn_apply<<<4096, 256, 0, stream>>>(x, stats, gamma, beta, ht,
                                       (long)NIMG * CH * hw);
  k_mb_transpose<<<64, 256, 0, stream>>>(mb, mbT);

  // 2) conv1x1 GEMMs (WMMA): theta, g over 14 images; phi over t=3 slices
  k_gemm<<<dim3(72, 4, NIMG), 256, 0, stream>>>(theta_w, ht, thetaT, theta_b,
        IC, CH, (long)CH * hw, (long)IC * hw, 1.f);
  k_gemm<<<dim3(72, 4, NIMG), 256, 0, stream>>>(g_w, ht, gT, g_b,
        IC, CH, (long)CH * hw, (long)IC * hw, 1.f);
  k_gemm<<<dim3(72, 4, NB), 256, 0, stream>>>(phi_w, ht + 3L * CH * hw, phiT,
        phi_b, IC, CH, (long)NT * CH * hw, (long)IC * hw, 1.f);

  // 3) neighborhood attention -> f in (t,p,c) order == corr's flat layout
  k_attn<<<dim3(36, NT, NB), 256, 0, stream>>>(phiT, thetaT, gT, f);

  // 4) W_y = wz_w(128x448) @ corr(448x9216)
  k_gemm<<<dim3(72, 8, NB), 256, 0, stream>>>(wz_w, f, Wy, wz_b,
        CH, NT * IC, (long)NT * IC * hw, (long)CH * hw, 1.f);

  // 5) memory bank: logitsT = 0.125 * mbT(256x64) @ phiT(64x9216); softmax;
  //    y1T = mb(64x256) @ attnT(256x9216)
  k_gemm<<<dim3(72, 16, NB), 256, 0, stream>>>(mbT, phiT, lgts, nullptr,
        256, IC, (long)IC * hw, 256L * hw, 0.125f);
  k_softmax256<<<dim3(36, NB), 256, 0, stream>>>(lgts);
  k_gemm<<<dim3(72, 4, NB), 256, 0, stream>>>(mb, lgts, y1T, nullptr,
        IC, 256, 256L * hw, (long)IC * hw, 1.f);

  // 6) W_y1 = wz1_w(128x64) @ y1T(64x9216)
  k_gemm<<<dim3(72, 8, NB), 256, 0, stream>>>(wz1_w, y1T, Wy1, wz1_b,
        CH, IC, (long)IC * hw, (long)CH * hw, 1.f);

  // 7) qloss (deterministic two-stage) + final residual sum into d_out
  k_l1_partial<<<1024, 256, 0, stream>>>(phiT, y1T, part, (long)NB * IC * hw);
  k_l1_final<<<1, 256, 0, stream>>>(part, 1024, out + 2359296L,
                                    1.f / (2.f * 64.f * 9216.f));
  k_final_add<<<4608, 256, 0, stream>>>(Wy, Wy1, x, out, (long)NB * CH * hw);
}